// ResidualVectorQuantizer_19258633355643
// MI455X (gfx1250) — compile-verified
//
#include <hip/hip_runtime.h>

#define DIM   256
#define CBSZ  1024
#define NCB   8
#define NB    16
#define NT    2000
#define TILE_M 32
#define THREADS 256
#define NWAVES 8
#define RSTRIDE 260   // 256 + 4 pad: 4-bank rotation per row -> conflict-free A reads

typedef float v2f __attribute__((ext_vector_type(2)));
typedef float v8f __attribute__((ext_vector_type(8)));

// ---- Pass 1: 0.5 * ||c||^2 per codebook entry -> d_ws ----
__global__ __launch_bounds__(THREADS)
void rvq_norms_kernel(const float* __restrict__ cb, float* __restrict__ norms) {
    int e = blockIdx.x * blockDim.x + threadIdx.x;          // 0 .. NCB*CBSZ-1
    const float* row = cb + (size_t)e * DIM;
    float s = 0.f;
#pragma unroll 8
    for (int d = 0; d < DIM; d += 4) {
        float4 v = *(const float4*)(row + d);
        s += v.x * v.x + v.y * v.y + v.z * v.z + v.w * v.w;
    }
    norms[e] = 0.5f * s;
}

// ---- Pass 2: full RVQ for a 32-token tile, all 8 codebooks on-chip ----
__global__ __launch_bounds__(THREADS)
void rvq_main_kernel(const float* __restrict__ x, const float* __restrict__ cb,
                     const float* __restrict__ norms, float* __restrict__ out) {
    __shared__ float R[TILE_M][RSTRIDE];    // residual
    __shared__ float Q[TILE_M][DIM];        // accumulated quantized
    __shared__ float red_val[NWAVES][TILE_M];
    __shared__ int   red_idx[NWAVES][TILE_M];
    __shared__ int   fin_idx[TILE_M];

    const int tid  = threadIdx.x;
    const int wave = tid >> 5;
    const int lane = tid & 31;
    const int tilesPerB = (NT + TILE_M - 1) / TILE_M;       // 63
    const int b  = blockIdx.x / tilesPerB;
    const int t0 = (blockIdx.x % tilesPerB) * TILE_M;

    // --- load x tile: R[m][d] = x[b][d][t0+m]; coalesced along t ---
    {
        const int m = lane;                 // token in tile
        const int t = t0 + m;
#pragma unroll 4
        for (int i = 0; i < DIM / NWAVES; ++i) {
            int d = wave * (DIM / NWAVES) + i;
            float v = 0.f;
            if (t < NT) v = x[((size_t)b * DIM + d) * NT + t];
            R[m][d] = v;
            Q[m][d] = 0.f;
        }
    }
    __syncthreads();

    const int halfsel = lane >> 4;          // 0 | 1 (k-half select)
    const int l15     = lane & 15;          // row / col within 16-tile
    float* out_codes = out + (size_t)NB * DIM * NT;

    for (int kcb = 0; kcb < NCB; ++kcb) {
        const float* cbk = cb + (size_t)kcb * CBSZ * DIM;
        const float* nrm = norms + kcb * CBSZ;

        float bestv0[8], bestv1[8];
        int   besti0[8], besti1[8];
#pragma unroll
        for (int v = 0; v < 8; ++v) {
            bestv0[v] = -3.4e38f; bestv1[v] = -3.4e38f;
            besti0[v] = 0;        besti1[v] = 0;
        }

        // each wave scores its 128-column stripe: 8 N-tiles of 16
        for (int nt = 0; nt < 8; ++nt) {
            const int colbase = wave * 128 + nt * 16;
            const int col = colbase + l15;
            v8f acc0 = {}; v8f acc1 = {};
            const float* bcol = cbk + (size_t)col * DIM + 2 * halfsel;
#pragma unroll 4
            for (int kk = 0; kk < DIM / 4; ++kk) {
                // B frag: B[k][n] = cb[n][k] ; lane: col=l15, k = 4*kk + 2*halfsel + {0,1}
                v2f bf = *(const v2f*)(bcol + kk * 4);
                // A frags from LDS: row=l15 (+16 for m-tile 1), same k pattern
                v2f a0 = *(const v2f*)(&R[l15][kk * 4 + 2 * halfsel]);
                v2f a1 = *(const v2f*)(&R[16 + l15][kk * 4 + 2 * halfsel]);
                acc0 = __builtin_amdgcn_wmma_f32_16x16x4_f32(false, a0, false, bf,
                                                             (short)0, acc0, false, false);
                acc1 = __builtin_amdgcn_wmma_f32_16x16x4_f32(false, a1, false, bf,
                                                             (short)0, acc1, false, false);
            }
            // score = dot - 0.5*||c||^2 ; argmax == argmin of distance
            float nv = nrm[col];
#pragma unroll
            for (int v = 0; v < 8; ++v) {
                float s0 = acc0[v] - nv;
                if (s0 > bestv0[v] || (s0 == bestv0[v] && col < besti0[v])) { bestv0[v] = s0; besti0[v] = col; }
                float s1 = acc1[v] - nv;
                if (s1 > bestv1[v] || (s1 == bestv1[v] && col < besti1[v])) { bestv1[v] = s1; besti1[v] = col; }
            }
        }

        // cross-lane reduce within each 16-lane half (same row set per half)
#pragma unroll
        for (int v = 0; v < 8; ++v) {
#pragma unroll
            for (int off = 8; off >= 1; off >>= 1) {
                float ov = __shfl_xor(bestv0[v], off, 32);
                int   oi = __shfl_xor(besti0[v], off, 32);
                if (ov > bestv0[v] || (ov == bestv0[v] && oi < besti0[v])) { bestv0[v] = ov; besti0[v] = oi; }
                ov = __shfl_xor(bestv1[v], off, 32);
                oi = __shfl_xor(besti1[v], off, 32);
                if (ov > bestv1[v] || (ov == bestv1[v] && oi < besti1[v])) { bestv1[v] = ov; besti1[v] = oi; }
            }
        }
        if (l15 == 0) {
#pragma unroll
            for (int v = 0; v < 8; ++v) {
                int r = v + 8 * halfsel;                 // rows 0-7 (lane0) / 8-15 (lane16)
                red_val[wave][r]      = bestv0[v];
                red_idx[wave][r]      = besti0[v];
                red_val[wave][16 + r] = bestv1[v];
                red_idx[wave][16 + r] = besti1[v];
            }
        }
        __syncthreads();

        // reduce across the 8 waves; write code output
        if (tid < TILE_M) {
            float bv = red_val[0][tid]; int bi = red_idx[0][tid];
#pragma unroll
            for (int w = 1; w < NWAVES; ++w) {
                float ov = red_val[w][tid]; int oi = red_idx[w][tid];
                if (ov > bv || (ov == bv && oi < bi)) { bv = ov; bi = oi; }
            }
            fin_idx[tid] = bi;
            int t = t0 + tid;
            if (t < NT) out_codes[((size_t)b * NCB + kcb) * NT + t] = (float)bi;
        }
        __syncthreads();

        // residual/quantized update: thread tid owns dim tid, coalesced gather
        for (int m = 0; m < TILE_M; ++m) {
            float c = cbk[(size_t)fin_idx[m] * DIM + tid];
            R[m][tid] -= c;
            Q[m][tid] += c;
        }
        __syncthreads();
    }

    // --- write quantized: out[b][d][t0+m] = Q[m][d] ---
    {
        const int m = lane;
        const int t = t0 + m;
        if (t < NT) {
#pragma unroll 4
            for (int i = 0; i < DIM / NWAVES; ++i) {
                int d = wave * (DIM / NWAVES) + i;
                out[((size_t)b * DIM + d) * NT + t] = Q[m][d];
            }
        }
    }
}

extern "C" void kernel_launch(void* const* d_in, const int* in_sizes, int n_in,
                              void* d_out, int out_size, void* d_ws, size_t ws_size,
                              hipStream_t stream) {
    (void)in_sizes; (void)n_in; (void)out_size; (void)ws_size;
    const float* x  = (const float*)d_in[0];   // (16, 256, 2000) f32
    const float* cb = (const float*)d_in[1];   // (8, 1024, 256) f32
    float* out   = (float*)d_out;              // quantized (16,256,2000) ++ codes (16,8,2000)
    float* norms = (float*)d_ws;               // 8*1024 floats of 0.5*||c||^2

    rvq_norms_kernel<<<(NCB * CBSZ) / THREADS, THREADS, 0, stream>>>(cb, norms);

    const int tilesPerB = (NT + TILE_M - 1) / TILE_M;   // 63
    rvq_main_kernel<<<NB * tilesPerB, THREADS, 0, stream>>>(x, cb, norms, out);
}